// FCOS_10866267259024
// MI455X (gfx1250) — compile-verified
//
#include <hip/hip_runtime.h>
#include <hip/hip_bf16.h>
#include <stdint.h>

// ---------------- problem constants (from the reference) ----------------
#define IMGSZ     2048.0f
#define NCLS      80
#define BATCH     8
#define TOPK      1000
#define NCAND     3000          // 3 levels * TOPK
#define NPAD      3008          // padded to 188*16 for 16x16 WMMA tiles
#define NWORDS    94            // 3008 bits / 32 = suppression-mask words per row
#define DETS      100
#define NBINS     2048
#define SCORE_TH  0.2f
#define NMS_TH    0.6f

typedef __attribute__((ext_vector_type(2))) float v2f;
typedef __attribute__((ext_vector_type(8))) float v8f;

// ---------------- workspace layout (4-byte units) ----------------
// [0,      24000)   selScore  float [img][3000]
// [24000, 120000)   selBox    float [img][3000][4]   (raw decoded+clipped)
// [120000,144000)   selLabel  int   [img][3000]
// [144000,168000)   oi        int   [img][3000]      (sorted order)
// [168000,192000)   ss        float [img][3000]      (sorted scores)
// [192000,2448000)  mask      uint  [img][3000][94]  (suppression bits)

// ======================================================================
// K0: zero the suppression mask (atomicOr target; must be zeroed per call)
// ======================================================================
__global__ void k_zero_mask(unsigned int* __restrict__ mask, int n) {
  for (int i = blockIdx.x * blockDim.x + threadIdx.x; i < n;
       i += gridDim.x * blockDim.x)
    mask[i] = 0u;
}

// ======================================================================
// K1: per (image, level) score -> histogram top-K select + box decode
// ======================================================================
__global__ __launch_bounds__(256) void k_topk(
    const float* __restrict__ cls0, const float* __restrict__ reg0,
    const float* __restrict__ ctr0, const float* __restrict__ anc0,
    const float* __restrict__ cls1, const float* __restrict__ reg1,
    const float* __restrict__ ctr1, const float* __restrict__ anc1,
    const float* __restrict__ cls2, const float* __restrict__ reg2,
    const float* __restrict__ ctr2, const float* __restrict__ anc2,
    float* __restrict__ selScore, float* __restrict__ selBox,
    int* __restrict__ selLabel)
{
  const int lvl = blockIdx.x % 3;
  const int img = blockIdx.x / 3;
  const int hw  = (lvl == 0) ? 4096 : (lvl == 1 ? 1024 : 256);
  const float* cls = (lvl == 0) ? cls0 : (lvl == 1 ? cls1 : cls2);
  const float* reg = (lvl == 0) ? reg0 : (lvl == 1 ? reg1 : reg2);
  const float* ctr = (lvl == 0) ? ctr0 : (lvl == 1 ? ctr1 : ctr2);
  const float* anc = (lvl == 0) ? anc0 : (lvl == 1 ? anc1 : anc2);
  cls += (size_t)img * hw * NCLS;
  reg += (size_t)img * hw * 4;
  ctr += (size_t)img * hw;

  __shared__ unsigned int hist[NBINS];
  __shared__ unsigned int cntA, cntB;
  __shared__ int sCut, sAbove, sNeed, sTotal;

  for (int b = threadIdx.x; b < NBINS; b += blockDim.x) hist[b] = 0u;
  if (threadIdx.x == 0) { cntA = 0u; cntB = 0u; }
  __syncthreads();

  const int total = hw * NCLS;
  const float binScale = (float)NBINS / (1.0f - SCORE_TH);

  // pass 1: histogram of scores above threshold
  for (int e = threadIdx.x; e < total; e += blockDim.x) {
    const int a = e / NCLS;
    const float sc  = 1.0f / (1.0f + __expf(-cls[e]));
    const float sct = 1.0f / (1.0f + __expf(-ctr[a]));
    const float s = sqrtf(sc * sct);
    if (s > SCORE_TH) {
      int b = (int)((s - SCORE_TH) * binScale);
      b = (b < 0) ? 0 : (b > NBINS - 1 ? NBINS - 1 : b);
      atomicAdd(&hist[b], 1u);
    }
  }
  __syncthreads();

  if (threadIdx.x == 0) {
    unsigned int fullSum = 0;
    for (int b = 0; b < NBINS; ++b) fullSum += hist[b];
    unsigned int cum = 0, ab = 0;
    int cb = -1;
    for (int b = NBINS - 1; b >= 0; --b) {
      const unsigned int h = hist[b];
      if (cum + h >= (unsigned)TOPK) { cb = b; ab = cum; break; }
      cum += h;
    }
    if (cb < 0) { cb = 0; ab = fullSum - hist[0]; }   // fewer than TOPK total
    sCut = cb; sAbove = (int)ab; sNeed = TOPK - (int)ab;
    sTotal = (fullSum < (unsigned)TOPK) ? (int)fullSum : TOPK;
  }
  __syncthreads();

  const int cut = sCut, above = sAbove, need = sNeed;
  float* sbB = selBox   + ((size_t)img * NCAND + lvl * TOPK) * 4;
  float* ssB = selScore +  (size_t)img * NCAND + lvl * TOPK;
  int*   slB = selLabel +  (size_t)img * NCAND + lvl * TOPK;

  // pass 2: compact winners, decode + clip boxes
  for (int e = threadIdx.x; e < total; e += blockDim.x) {
    const int a   = e / NCLS;
    const int lab = e - a * NCLS;
    const float sc  = 1.0f / (1.0f + __expf(-cls[e]));
    const float sct = 1.0f / (1.0f + __expf(-ctr[a]));
    const float s = sqrtf(sc * sct);
    if (s <= SCORE_TH) continue;
    int b = (int)((s - SCORE_TH) * binScale);
    b = (b < 0) ? 0 : (b > NBINS - 1 ? NBINS - 1 : b);
    int pos = -1;
    if (b > cut) {
      pos = (int)atomicAdd(&cntA, 1u);
    } else if (b == cut) {
      const int q = (int)atomicAdd(&cntB, 1u);
      if (q < need) pos = above + q;
    }
    if (pos < 0 || pos >= TOPK) continue;

    const float a0 = anc[a * 4 + 0], a1 = anc[a * 4 + 1];
    const float a2 = anc[a * 4 + 2], a3 = anc[a * 4 + 3];
    const float cx = 0.5f * (a0 + a2), cy = 0.5f * (a1 + a3);
    const float w = a2 - a0, h = a3 - a1;
    const float r0 = reg[a * 4 + 0] * w, r1 = reg[a * 4 + 1] * h;
    const float r2 = reg[a * 4 + 2] * w, r3 = reg[a * 4 + 3] * h;
    const float x1 = fminf(fmaxf(cx - r0, 0.f), IMGSZ);
    const float y1 = fminf(fmaxf(cy - r1, 0.f), IMGSZ);
    const float x2 = fminf(fmaxf(cx + r2, 0.f), IMGSZ);
    const float y2 = fminf(fmaxf(cy + r3, 0.f), IMGSZ);
    ssB[pos] = s;
    slB[pos] = lab;
    sbB[pos * 4 + 0] = x1; sbB[pos * 4 + 1] = y1;
    sbB[pos * 4 + 2] = x2; sbB[pos * 4 + 3] = y2;
  }
  __syncthreads();

  // pass 3: pad remaining slots (score=-1 marks padding, like the reference)
  for (int p = sTotal + (int)threadIdx.x; p < TOPK; p += blockDim.x) {
    ssB[p] = -1.0f; slB[p] = 0;
    sbB[p * 4 + 0] = 0.f; sbB[p * 4 + 1] = 0.f;
    sbB[p * 4 + 2] = 0.f; sbB[p * 4 + 3] = 0.f;
  }
}

// ======================================================================
// K2: per image bitonic sort (descending score) of 3000 candidates
// ======================================================================
#define SORTN 4096
__global__ __launch_bounds__(1024) void k_sort(
    const float* __restrict__ selScore,
    int* __restrict__ oi, float* __restrict__ ss)
{
  const int img = blockIdx.x;
  __shared__ float key[SORTN];
  __shared__ int   val[SORTN];
  const float* s = selScore + (size_t)img * NCAND;

  for (int e = threadIdx.x; e < SORTN; e += blockDim.x) {
    key[e] = (e < NCAND) ? s[e] : -2.0f;
    val[e] = (e < NCAND) ? e : 0;
  }
  __syncthreads();

  for (int k = 2; k <= SORTN; k <<= 1) {
    for (int j = k >> 1; j > 0; j >>= 1) {
      for (int e = threadIdx.x; e < SORTN; e += blockDim.x) {
        const int x = e ^ j;
        if (x > e) {
          const bool desc = ((e & k) == 0);
          const float ke = key[e], kx = key[x];
          if ((ke < kx) == desc) {
            key[e] = kx; key[x] = ke;
            const int t = val[e]; val[e] = val[x]; val[x] = t;
          }
        }
      }
      __syncthreads();
    }
  }

  for (int e = threadIdx.x; e < NCAND; e += blockDim.x) {
    oi[(size_t)img * NCAND + e] = val[e];
    ss[(size_t)img * NCAND + e] = key[e];
  }
}

// ======================================================================
// K3: per image suppression-mask build.
//  - async gather sorted boxes into LDS (global_load_async_to_lds_b128)
//  - one wave per 16x16 tile: WMMA outer product gives area_i + area_j,
//    VALU computes intersection, atomicOr writes suppression bits (j > i).
// ======================================================================
__global__ __launch_bounds__(1024) void k_iou(
    const float* __restrict__ selBox, const int* __restrict__ selLabel,
    const int* __restrict__ oi, unsigned int* __restrict__ mask)
{
  const int img = blockIdx.x;
  __shared__ float sbox[NPAD * 4];   // class-offset boxes, sorted order
  __shared__ float sarea[NPAD];
  const int* oiI = oi + (size_t)img * NCAND;
  const unsigned long long boxBase =
      (unsigned long long)(uintptr_t)selBox + (unsigned long long)img * NCAND * 16ull;

  // async per-lane gather: global boxes[oi[j]] -> LDS sbox[j]
  for (int jj = threadIdx.x; jj < NCAND; jj += blockDim.x) {
    const int orig = oiI[jj];
    const unsigned long long g = boxBase + (unsigned long long)orig * 16ull;
    const unsigned int l = (unsigned int)(uintptr_t)(&sbox[jj * 4]);
    asm volatile("global_load_async_to_lds_b128 %0, %1, off"
                 :: "v"(l), "v"(g) : "memory");
  }
  for (int jj = NCAND + (int)threadIdx.x; jj < NPAD; jj += blockDim.x) {
    sbox[jj * 4 + 0] = 0.f; sbox[jj * 4 + 1] = 0.f;
    sbox[jj * 4 + 2] = 0.f; sbox[jj * 4 + 3] = 0.f;
  }
  asm volatile("s_wait_asynccnt 0" ::: "memory");
  __syncthreads();

  // add per-class coordinate offsets (batched NMS) and compute areas
  for (int jj = threadIdx.x; jj < NPAD; jj += blockDim.x) {
    float off = 0.f;
    if (jj < NCAND)
      off = (float)selLabel[(size_t)img * NCAND + oiI[jj]] * (IMGSZ + 1.0f);
    const float x1 = sbox[jj * 4 + 0] + off, y1 = sbox[jj * 4 + 1] + off;
    const float x2 = sbox[jj * 4 + 2] + off, y2 = sbox[jj * 4 + 3] + off;
    sbox[jj * 4 + 0] = x1; sbox[jj * 4 + 1] = y1;
    sbox[jj * 4 + 2] = x2; sbox[jj * 4 + 3] = y2;
    sarea[jj] = (x2 - x1) * (y2 - y1);
  }
  __syncthreads();

  unsigned int* maskI = mask + (size_t)img * NCAND * NWORDS;
  const int wave   = threadIdx.x >> 5;
  const int lane   = threadIdx.x & 31;
  const int nwaves = blockDim.x >> 5;
  const int NT = NPAD / 16;   // 188 tiles per dimension

  for (int t = wave; t < NT * NT; t += nwaves) {
    const int ti = t / NT, tj = t % NT;
    if (tj < ti) continue;                 // only j > i region (wave-uniform)
    const int i0 = ti * 16, j0 = tj * 16;
    const int l15 = lane & 15;

    // A = [area_i, 1, 0, 0] (16x4), B = [1; area_j; 0; 0] (4x16):
    // D[m][n] = area_i[m] + area_j[n]  via one v_wmma_f32_16x16x4_f32
    v2f a, b;
    a.x = (lane < 16) ? sarea[i0 + l15] : 0.0f;
    a.y = (lane < 16) ? 1.0f : 0.0f;
    b.x = (lane < 16) ? 1.0f : 0.0f;
    b.y = (lane < 16) ? sarea[j0 + l15] : 0.0f;
    v8f c = {};
    v8f d = __builtin_amdgcn_wmma_f32_16x16x4_f32(
        false, a, false, b, (short)0, c, false, false);

    const int n = j0 + l15;                // column handled by this lane
    const float bx1 = sbox[n * 4 + 0], by1 = sbox[n * 4 + 1];
    const float bx2 = sbox[n * 4 + 2], by2 = sbox[n * 4 + 3];
    const int mb = (lane < 16) ? 0 : 8;    // D rows: lanes<16 -> M 0..7, else 8..15
#pragma unroll
    for (int r = 0; r < 8; ++r) {
      const int m = i0 + mb + r;
      const float ax1 = sbox[m * 4 + 0], ay1 = sbox[m * 4 + 1];
      const float ax2 = sbox[m * 4 + 2], ay2 = sbox[m * 4 + 3];
      const float ix1 = fmaxf(ax1, bx1), iy1 = fmaxf(ay1, by1);
      const float ix2 = fminf(ax2, bx2), iy2 = fminf(ay2, by2);
      const float iw = fmaxf(ix2 - ix1, 0.f), ih = fmaxf(iy2 - iy1, 0.f);
      const float inter = iw * ih;
      const float uni = d[r] - inter;      // area_i + area_j - inter (WMMA term)
      const bool sup = inter > NMS_TH * uni;
      if (sup && n > m && m < NCAND && n < NCAND)
        atomicOr(&maskI[(size_t)m * NWORDS + (n >> 5)], 1u << (n & 31));
    }
  }
}

// ======================================================================
// K4: per image greedy keep over the bitmask + emit top-100 detections
// ======================================================================
__global__ __launch_bounds__(128) void k_nms_out(
    const float* __restrict__ selBox, const int* __restrict__ selLabel,
    const int* __restrict__ oi, const float* __restrict__ ss,
    const unsigned int* __restrict__ mask,
    float* __restrict__ outBoxes, float* __restrict__ outScores,
    float* __restrict__ outLabels)
{
  const int img = blockIdx.x;
  __shared__ unsigned int rem[NWORDS];
  __shared__ int list[DETS];
  __shared__ int cnt;
  __shared__ int isKept;

  for (int w = threadIdx.x; w < NWORDS; w += blockDim.x) rem[w] = 0u;
  if (threadIdx.x == 0) cnt = 0;
  __syncthreads();

  const float* ssI = ss + (size_t)img * NCAND;
  const unsigned int* maskI = mask + (size_t)img * NCAND * NWORDS;

  for (int i = 0; i < NCAND; ++i) {
    if (threadIdx.x == 0) {
      const bool valid  = ssI[i] > 0.0f;
      const bool notrem = ((rem[i >> 5] >> (i & 31)) & 1u) == 0u;
      const int k = (valid && notrem) ? 1 : 0;
      if (k && cnt < DETS) list[cnt] = i;
      if (k) cnt++;
      isKept = k;
    }
    __syncthreads();
    if (isKept) {
      for (int w = threadIdx.x; w < NWORDS; w += blockDim.x)
        rem[w] |= maskI[(size_t)i * NWORDS + w];
    }
    __syncthreads();
  }

  const int nk = (cnt < DETS) ? cnt : DETS;
  const int* oiI = oi + (size_t)img * NCAND;
  for (int d = threadIdx.x; d < DETS; d += blockDim.x) {
    float b0 = 0.f, b1 = 0.f, b2 = 0.f, b3 = 0.f, sc = -1.0f, lb = 0.0f;
    if (d < nk) {
      const int i = list[d];
      const int orig = oiI[i];
      const float* bp = selBox + ((size_t)img * NCAND + orig) * 4;
      b0 = bp[0]; b1 = bp[1]; b2 = bp[2]; b3 = bp[3];
      sc = ssI[i];
      lb = (float)selLabel[(size_t)img * NCAND + orig];
    }
    outBoxes[((size_t)img * DETS + d) * 4 + 0] = b0;
    outBoxes[((size_t)img * DETS + d) * 4 + 1] = b1;
    outBoxes[((size_t)img * DETS + d) * 4 + 2] = b2;
    outBoxes[((size_t)img * DETS + d) * 4 + 3] = b3;
    outScores[(size_t)img * DETS + d] = sc;
    outLabels[(size_t)img * DETS + d] = lb;
  }
}

// ======================================================================
extern "C" void kernel_launch(void* const* d_in, const int* in_sizes, int n_in,
                              void* d_out, int out_size, void* d_ws, size_t ws_size,
                              hipStream_t stream)
{
  (void)in_sizes; (void)n_in; (void)out_size; (void)ws_size;
  const float* cls0 = (const float*)d_in[0];
  const float* reg0 = (const float*)d_in[1];
  const float* ctr0 = (const float*)d_in[2];
  const float* anc0 = (const float*)d_in[3];
  const float* cls1 = (const float*)d_in[4];
  const float* reg1 = (const float*)d_in[5];
  const float* ctr1 = (const float*)d_in[6];
  const float* anc1 = (const float*)d_in[7];
  const float* cls2 = (const float*)d_in[8];
  const float* reg2 = (const float*)d_in[9];
  const float* ctr2 = (const float*)d_in[10];
  const float* anc2 = (const float*)d_in[11];

  float* ws = (float*)d_ws;
  float* selScore = ws;                           // 24000 f32
  float* selBox   = ws + 24000;                   // 96000 f32
  int*   selLabel = (int*)(ws + 120000);          // 24000 i32
  int*   oiArr    = (int*)(ws + 144000);          // 24000 i32
  float* ssArr    = ws + 168000;                  // 24000 f32
  unsigned int* mask = (unsigned int*)(ws + 192000); // 2,256,000 u32

  float* outBoxes  = (float*)d_out;               // [8][100][4]
  float* outScores = outBoxes + BATCH * DETS * 4; // [8][100]
  float* outLabels = outScores + BATCH * DETS;    // [8][100]

  const int maskWords = BATCH * NCAND * NWORDS;
  k_zero_mask<<<1024, 256, 0, stream>>>(mask, maskWords);
  k_topk<<<BATCH * 3, 256, 0, stream>>>(cls0, reg0, ctr0, anc0,
                                        cls1, reg1, ctr1, anc1,
                                        cls2, reg2, ctr2, anc2,
                                        selScore, selBox, selLabel);
  k_sort<<<BATCH, 1024, 0, stream>>>(selScore, oiArr, ssArr);
  k_iou<<<BATCH, 1024, 0, stream>>>(selBox, selLabel, oiArr, mask);
  k_nms_out<<<BATCH, 128, 0, stream>>>(selBox, selLabel, oiArr, ssArr, mask,
                                       outBoxes, outScores, outLabels);
}